// TensorProductUniform1d_4621384810537
// MI455X (gfx1250) — compile-verified
//
#include <hip/hip_runtime.h>

typedef float v2f __attribute__((ext_vector_type(2)));
typedef float v8f __attribute__((ext_vector_type(8)));

#define TP_B  8192
#define TP_U  128
#define TP_S1 8
#define TP_S2 8
#define TP_SO 16
#define TP_P  128

// ---------------------------------------------------------------------------
// Setup kernel: densify the sparse path list into W2[o][s1*8+s2] (16 x 64).
// 1024 threads, one per (o,k) cell; each scans the 128 paths (hot in L2/L0).
// Deterministic, no atomics.
// ---------------------------------------------------------------------------
__global__ void tp_build_w2(const int* __restrict__ pidx,
                            const float* __restrict__ pcoef,
                            float* __restrict__ W2) {
    const int t  = threadIdx.x;          // 0..1023
    const int o  = t >> 6;               // 0..15
    const int k  = t & 63;               // 0..63
    const int s1 = k >> 3;
    const int s2 = k & 7;
    float sum = 0.0f;
    for (int p = 0; p < TP_P; ++p) {
        const int i1 = pidx[3 * p + 0];
        const int i2 = pidx[3 * p + 1];
        const int io = pidx[3 * p + 2];
        sum += (i1 == s1 && i2 == s2 && io == o) ? pcoef[p] : 0.0f;
    }
    W2[t] = sum;   // row-major: W2[o*64 + k]
}

// ---------------------------------------------------------------------------
// Main kernel: one wave32 per batch element b.
//   Out(16 x 128) = W2(16 x 64) @ Z(64 x 128), Z[k,u] = x1[b,k/8,u]*x2[b,k%8,u]
// processed as 8 column tiles of 16 u each, K=64 as 16 chained
// V_WMMA_F32_16X16X4_F32 ops.
//
// f32 WMMA operand layout (wave32):
//   A 16x4 : VGPR0 = {lanes0-15: K=0, lanes16-31: K=2}, VGPR1 = {K=1, K=3}
//   B 4x16 : VGPR0 = {lanes0-15: K=0, lanes16-31: K=2}, VGPR1 = {K=1, K=3}
//   C 16x16: VGPR i = {lanes0-15: M=i, lanes16-31: M=i+8}, N = lane%16
// so lane (h = lane>>4, m = lane&15) supplies, for K-chunk j:
//   A[j] = { W2[m][4j+2h], W2[m][4j+2h+1] }
//   B[j] = { Z [4j+2h][m], Z [4j+2h+1][m] }
//
// The 2h lane-half dependence is folded into LOAD ADDRESSES (not register
// indices) so all register indexing is compile-time static: no v_cndmask,
// and each lane loads only the 4 x2 rows its half consumes.
//   k = 4j+2h+d  ->  s1 = j>>1 (all lanes),  s2 = (j&1)*4 + 2h + d
//   lane-local x2 rows needed: {2h, 2h+1, 4+2h, 5+2h}  -> q[0..3]
// ---------------------------------------------------------------------------
__global__ void __launch_bounds__(256)
tp_main(const float* __restrict__ x1,
        const float* __restrict__ x2,
        const float* __restrict__ W2,
        float* __restrict__ out) {
    const int wave = (int)((blockIdx.x * blockDim.x + threadIdx.x) >> 5);
    const int lane = (int)(threadIdx.x & 31);
    const int b    = wave;                 // one wave per batch element
    const int hh   = lane >> 4;            // lane half: K parity pair select
    const int m    = lane & 15;            // row (for A) / column (for B,D)

    // --- A operands: W2 chunks, loaded once per wave, reused for 8 tiles ---
    // half-dependent offset folded into the base pointer (static thereafter)
    const float* W2l = W2 + m * 64 + 2 * hh;
    v2f A[16];
#pragma unroll
    for (int j = 0; j < 16; ++j) {
        A[j] = *(const v2f*)(W2l + 4 * j);   // {W2[m][4j+2h], W2[m][4j+2h+1]}
    }

    const float* x1b = x1 + (size_t)b * TP_S1 * TP_U;
    const float* x2l = x2 + (size_t)b * TP_S2 * TP_U + (2 * hh) * TP_U;
    float*       ob  = out + (size_t)b * TP_SO * TP_U + (8 * hh) * TP_U;

#pragma unroll
    for (int t = 0; t < 8; ++t) {                 // 8 tiles of 16 columns (u)
        const int u = t * 16 + m;

        // x1: all 8 rows (needed by both halves; same cache lines -> no
        // extra HBM traffic). x2: only this half's 4 rows.
        float a1[TP_S1];
#pragma unroll
        for (int s = 0; s < TP_S1; ++s) a1[s] = x1b[s * TP_U + u];
        float q[4];
        q[0] = x2l[0 * TP_U + u];   // s2 = 2h
        q[1] = x2l[1 * TP_U + u];   // s2 = 2h+1
        q[2] = x2l[4 * TP_U + u];   // s2 = 4+2h
        q[3] = x2l[5 * TP_U + u];   // s2 = 5+2h

        // K=64 as 16 chained f32 WMMAs, accumulator starts at 0.
        v8f c = {0.f, 0.f, 0.f, 0.f, 0.f, 0.f, 0.f, 0.f};
#pragma unroll
        for (int j = 0; j < 16; ++j) {
            const int s1 = j >> 1;            // static
            const int qb = (j & 1) * 2;       // static
            v2f bo;
            bo.x = a1[s1] * q[qb];            // Z[4j+2h  ][m]
            bo.y = a1[s1] * q[qb + 1];        // Z[4j+2h+1][m]
            c = __builtin_amdgcn_wmma_f32_16x16x4_f32(
                    /*neg_a=*/false, A[j],
                    /*neg_b=*/false, bo,
                    /*c_mod=*/(short)0, c,
                    /*reuse_a=*/false, /*reuse_b=*/false);
        }

        // D tile: lane half h holds output rows i+8h, i = 0..7, column u.
#pragma unroll
        for (int i = 0; i < 8; ++i) {
            ob[i * TP_U + u] = c[i];
        }
    }
}

// ---------------------------------------------------------------------------
// Inputs (setup_inputs order): x1 f32[B,S1,U], x2 f32[B,S2,U],
// path_indices i32[P,3], path_coefficients f32[P].
// Output: reference returns shape (B, SO, U); our layout
// out[b*SO*U + o*U + u] matches.
// ---------------------------------------------------------------------------
extern "C" void kernel_launch(void* const* d_in, const int* in_sizes, int n_in,
                              void* d_out, int out_size, void* d_ws, size_t ws_size,
                              hipStream_t stream) {
    const float* x1    = (const float*)d_in[0];
    const float* x2    = (const float*)d_in[1];
    const int*   pidx  = (const int*)  d_in[2];
    const float* pcoef = (const float*)d_in[3];
    float*       out   = (float*)d_out;
    float*       W2    = (float*)d_ws;            // 16*64 floats = 4 KB scratch

    const int B = in_sizes[0] / (TP_S1 * TP_U);   // 8192

    tp_build_w2<<<1, 1024, 0, stream>>>(pidx, pcoef, W2);

    // one wave32 per b: B*32 threads, 256-thread blocks (8 waves/block)
    const int threads = 256;
    const int blocks  = (B * 32) / threads;
    tp_main<<<blocks, threads, 0, stream>>>(x1, x2, W2, out);
}